// Decoder_66425964200038
// MI455X (gfx1250) — compile-verified
//
#include <hip/hip_runtime.h>
#include <math.h>

typedef __attribute__((ext_vector_type(2))) float v2f;
typedef __attribute__((ext_vector_type(8))) float v8f;

#define BB    4
#define CHN   64
#define EMB   32
#define TFULL 4096

__device__ __forceinline__ float gelu_exact(float x) {
  return 0.5f * x * (1.0f + erff(x * 0.7071067811865475f));
}

// ------------------------------------------------------------------
// 1x1 conv as WMMA f32 GEMM.
// Y[b,o,t] = (sum_c W[o,c]*(a1*X1[b,c,t]+a2*X2[b,c,t]) + bias[o] (+res[b,o,t])) * mask[b,t]
// One wave computes a (MT*16 x 16) output tile. K stepped by 4 (V_WMMA_F32_16X16X4_F32).
// A frag (16x4): lane l, vgpr j -> A[M=l&15, K=2*(l>>4)+j]   (b64 vector load)
// B frag (4x16): lane l, vgpr j -> B[K=2*(l>>4)+j, N=l&15]
// C/D (16x16):   vgpr r, lane l -> C[M=r+8*(l>>4), N=l&15]
// ------------------------------------------------------------------
template <int MT>
__global__ void pw_conv_wmma(const float* __restrict__ W,
                             const float* __restrict__ bias,
                             const float* __restrict__ X1,
                             const float* __restrict__ X2,
                             float a1, float a2,
                             const float* __restrict__ res,
                             const float* __restrict__ mrow,
                             float* __restrict__ Y,
                             int Cin, int L)
{
  const int lane = threadIdx.x & 31;
  const int wave = threadIdx.x >> 5;
  const int b    = blockIdx.y;
  const int t0   = (blockIdx.x * (blockDim.x >> 5) + wave) * 16;
  if (t0 >= L) return;
  const int half = lane >> 4;
  const int ln   = lane & 15;

  const float* Xb1 = X1 + (size_t)b * Cin * L;
  const float* Xb2 = X2 ? (X2 + (size_t)b * Cin * L) : nullptr;

  const v8f zero = {0.f,0.f,0.f,0.f,0.f,0.f,0.f,0.f};
  v8f acc[MT];
#pragma unroll
  for (int m = 0; m < MT; ++m) acc[m] = zero;

  for (int k = 0; k < Cin; k += 4) {
    const int kr = k + 2 * half;
    v2f bf;
    float x0 = a1 * Xb1[(size_t)kr * L + t0 + ln];
    float x1 = a1 * Xb1[(size_t)(kr + 1) * L + t0 + ln];
    if (Xb2) {
      x0 += a2 * Xb2[(size_t)kr * L + t0 + ln];
      x1 += a2 * Xb2[(size_t)(kr + 1) * L + t0 + ln];
    }
    bf[0] = x0; bf[1] = x1;
#pragma unroll
    for (int m = 0; m < MT; ++m) {
      // contiguous, 8B-aligned pair -> single global_load_b64
      const v2f af = *reinterpret_cast<const v2f*>(W + (size_t)(m * 16 + ln) * Cin + kr);
      acc[m] = __builtin_amdgcn_wmma_f32_16x16x4_f32(false, af, false, bf,
                                                     (short)0, acc[m], false, false);
    }
  }

  const float mval = mrow ? mrow[(size_t)b * L + t0 + ln] : 1.0f;
#pragma unroll
  for (int m = 0; m < MT; ++m) {
#pragma unroll
    for (int r = 0; r < 8; ++r) {
      const int o = m * 16 + r + 8 * half;
      float val = acc[m][r] + bias[o];
      if (res) val += res[((size_t)b * (MT * 16) + o) * L + t0 + ln];
      val *= mval;
      Y[((size_t)b * (MT * 16) + o) * L + t0 + ln] = val;
    }
  }
}

// Repack ff_w [o][c][s] -> [s][o][c] so dconv A-fragments are contiguous pairs.
__global__ void repack_w3_kernel(const float* __restrict__ W3, float* __restrict__ Wr)
{
  int i = blockIdx.x * blockDim.x + threadIdx.x;
  if (i >= 3 * CHN * CHN) return;
  int c = i & 63, o = (i >> 6) & 63, s = i >> 12;
  Wr[i] = W3[(o * CHN + c) * 3 + s];
}

// ------------------------------------------------------------------
// Dilated 3-tap conv (padding=dil) + exact GELU, via 3 accumulated WMMA GEMMs.
// Wr layout [3][64][64] (repacked). Branchless boundary handling:
// clamp index + multiply by hoisted 0/1 lane mask (no exec divergence in K loop).
// ------------------------------------------------------------------
__global__ void dconv3_gelu_wmma(const float* __restrict__ Wr,
                                 const float* __restrict__ bias,
                                 const float* __restrict__ X,
                                 float* __restrict__ Y,
                                 int L, int dil)
{
  const int lane = threadIdx.x & 31;
  const int wave = threadIdx.x >> 5;
  const int b    = blockIdx.y;
  const int t0   = (blockIdx.x * (blockDim.x >> 5) + wave) * 16;
  if (t0 >= L) return;
  const int half = lane >> 4;
  const int ln   = lane & 15;
  const float* Xb = X + (size_t)b * CHN * L;

  const v8f zero = {0.f,0.f,0.f,0.f,0.f,0.f,0.f,0.f};
  v8f acc[4];
#pragma unroll
  for (int m = 0; m < 4; ++m) acc[m] = zero;

#pragma unroll
  for (int s = 0; s < 3; ++s) {
    const int tt = t0 + ln + (s - 1) * dil;
    const float bmask = (tt >= 0 && tt < L) ? 1.0f : 0.0f;
    const int ttc = min(max(tt, 0), L - 1);
    const float* Ws = Wr + (size_t)s * CHN * CHN;
    for (int k = 0; k < CHN; k += 4) {
      const int kr = k + 2 * half;
      v2f bf;
      bf[0] = bmask * Xb[(size_t)kr * L + ttc];
      bf[1] = bmask * Xb[(size_t)(kr + 1) * L + ttc];
#pragma unroll
      for (int m = 0; m < 4; ++m) {
        const v2f af = *reinterpret_cast<const v2f*>(Ws + (size_t)(m * 16 + ln) * CHN + kr);
        acc[m] = __builtin_amdgcn_wmma_f32_16x16x4_f32(false, af, false, bf,
                                                       (short)0, acc[m], false, false);
      }
    }
  }

#pragma unroll
  for (int m = 0; m < 4; ++m) {
#pragma unroll
    for (int r = 0; r < 8; ++r) {
      const int o = m * 16 + r + 8 * half;
      float val = gelu_exact(acc[m][r] + bias[o]);
      Y[((size_t)b * CHN + o) * L + t0 + ln] = val;
    }
  }
}

// ------------------------------------------------------------------
// Per-channel BN statistics over (B, L): mu[c], rs[c] = rsqrt(var+1e-5)
// ------------------------------------------------------------------
__global__ void bn_stats_kernel(const float* __restrict__ ff, int L,
                                float* __restrict__ mu, float* __restrict__ rs)
{
  const int c = blockIdx.x;
  __shared__ float s1[256], s2[256];
  float a = 0.f, q = 0.f;
  for (int b = 0; b < BB; ++b) {
    const float* p = ff + ((size_t)b * CHN + c) * L;
    for (int t = threadIdx.x; t < L; t += blockDim.x) {
      float v = p[t]; a += v; q += v * v;
    }
  }
  s1[threadIdx.x] = a; s2[threadIdx.x] = q;
  __syncthreads();
  for (int off = 128; off > 0; off >>= 1) {
    if ((int)threadIdx.x < off) {
      s1[threadIdx.x] += s1[threadIdx.x + off];
      s2[threadIdx.x] += s2[threadIdx.x + off];
    }
    __syncthreads();
  }
  if (threadIdx.x == 0) {
    float n = (float)(BB * L);
    float m = s1[0] / n;
    float var = s2[0] / n - m * m;
    mu[c] = m;
    rs[c] = rsqrtf(var + 1e-5f);
  }
}

// Fold BN into q-conv: qw_eff[o,c]=qw*g*rs ; qb_eff[o]=qb+sum_c qw*(beta-mu*g*rs)
__global__ void fold_bn_kernel(const float* __restrict__ qw, const float* __restrict__ qb,
                               const float* __restrict__ g,  const float* __restrict__ beta,
                               const float* __restrict__ mu, const float* __restrict__ rs,
                               float* __restrict__ qw_eff,   float* __restrict__ qb_eff)
{
  for (int i = threadIdx.x; i < EMB * CHN; i += blockDim.x) {
    int c = i & (CHN - 1);
    qw_eff[i] = qw[i] * g[c] * rs[c];
  }
  if (threadIdx.x < EMB) {
    int o = threadIdx.x;
    float acc = qb[o];
    for (int c = 0; c < CHN; ++c)
      acc += qw[o * CHN + c] * (beta[c] - mu[c] * g[c] * rs[c]);
    qb_eff[o] = acc;
  }
}

// ------------------------------------------------------------------
// Sliding cross-attention, one thread per query position, online softmax.
// e = q.k/sqrt(32) + log(fm+1e-6); att = softmax(e)*fm; out = att @ v^T
// fm[j] = (j < KVd+Qd-1) * mp ; mp = (t >= Qd/2) ? mask[t-Qd/2] : 0
// Writes g = GELU(out)  (B, 32, L).
// ------------------------------------------------------------------
__global__ void sliding_att_kernel(const float* __restrict__ q,
                                   const float* __restrict__ k,
                                   const float* __restrict__ v,
                                   const float* __restrict__ mrow,
                                   float* __restrict__ g,
                                   int L, int Qd, int KVd)
{
  const int t = blockIdx.x * blockDim.x + threadIdx.x;
  const int b = blockIdx.y;
  if (t >= L) return;
  const int n = t / Qd;
  const float mpq = (t >= (Qd >> 1)) ? mrow[(size_t)b * L + t - (Qd >> 1)] : 0.0f;

  const float* qp = q + (size_t)b * EMB * L;
  const float* kp = k + (size_t)b * EMB * TFULL;
  const float* vp = v + (size_t)b * EMB * TFULL;

  float qr[EMB];
#pragma unroll
  for (int c = 0; c < EMB; ++c) qr[c] = qp[(size_t)c * L + t];
  float oacc[EMB];
#pragma unroll
  for (int c = 0; c < EMB; ++c) oacc[c] = 0.0f;

  float m = -3.0e38f, s = 0.0f;
  const float inv = 0.1767766952966369f; // 1/sqrt(32)
  const int tk0 = n * KVd - (KVd >> 1);
  const int wlim = KVd + Qd - 1;

  for (int j = 0; j < 2 * KVd; ++j) {
    const int tk = tk0 + j;
    const bool ok = (tk >= 0) && (tk < TFULL);
    float dot = 0.0f;
    if (ok) {
#pragma unroll
      for (int c = 0; c < EMB; ++c) dot += qr[c] * kp[(size_t)c * TFULL + tk];
    }
    const float fm = (j < wlim) ? mpq : 0.0f;
    const float e = dot * inv + logf(fm + 1e-6f);
    const float nm = fmaxf(m, e);
    const float sc = expf(m - nm);
    const float p  = expf(e - nm);
    s = s * sc + p;
    if (sc != 1.0f) {
#pragma unroll
      for (int c = 0; c < EMB; ++c) oacc[c] *= sc;
    }
    const float w = p * fm;
    if (ok && w != 0.0f) {
#pragma unroll
      for (int c = 0; c < EMB; ++c) oacc[c] += w * vp[(size_t)c * TFULL + tk];
    }
    m = nm;
  }

  const float invs = 1.0f / s;
  float* gp = g + (size_t)b * EMB * L;
#pragma unroll
  for (int c = 0; c < EMB; ++c) gp[(size_t)c * L + t] = gelu_exact(oacc[c] * invs);
}

// ------------------------------------------------------------------
// Elementwise resampling kernels (rows of length Lout).
// ------------------------------------------------------------------
__global__ void maxpool2_kernel(const float* __restrict__ X, float* __restrict__ Y,
                                int n, int Lout)
{
  int i = blockIdx.x * blockDim.x + threadIdx.x;
  if (i >= n) return;
  int t = i % Lout, r = i / Lout;
  const float* p = X + (size_t)r * (2 * Lout);
  Y[i] = fmaxf(p[2 * t], p[2 * t + 1]);
}

__global__ void rowsub2_kernel(const float* __restrict__ X, float* __restrict__ Y,
                               int n, int Lout)
{
  int i = blockIdx.x * blockDim.x + threadIdx.x;
  if (i >= n) return;
  int t = i % Lout, r = i / Lout;
  Y[i] = X[(size_t)r * (2 * Lout) + 2 * t];
}

__global__ void rowup2_kernel(const float* __restrict__ X, float* __restrict__ Y,
                              int n, int Lout)
{
  int i = blockIdx.x * blockDim.x + threadIdx.x;
  if (i >= n) return;
  int t = i % Lout, r = i / Lout;
  Y[i] = X[(size_t)r * (Lout >> 1) + (t >> 1)];
}

// ------------------------------------------------------------------
extern "C" void kernel_launch(void* const* d_in, const int* in_sizes, int n_in,
                              void* d_out, int out_size, void* d_ws, size_t ws_size,
                              hipStream_t stream)
{
  (void)in_sizes; (void)n_in; (void)out_size; (void)ws_size;

  auto F = [&](int i) -> const float* { return (const float*)d_in[i]; };

  struct Layer {
    const float *ao_b, *ao_w, *bn_b, *bn_g, *c1_b, *c1_w, *ff_b, *ff_w,
                *k_b, *k_w, *q_b, *q_w, *v_b, *v_w;
  };
  auto getL = [&](int s) -> Layer {
    Layer l;
    l.ao_b = F(s + 0);  l.ao_w = F(s + 1);
    l.bn_b = F(s + 2);  l.bn_g = F(s + 3);
    l.c1_b = F(s + 4);  l.c1_w = F(s + 5);
    l.ff_b = F(s + 6);  l.ff_w = F(s + 7);
    l.k_b  = F(s + 8);  l.k_w  = F(s + 9);
    l.q_b  = F(s + 10); l.q_w  = F(s + 11);
    l.v_b  = F(s + 12); l.v_w  = F(s + 13);
    return l;
  };

  // d_in layout: 0=x, 1=fencoder, 2=mask, then params flattened (sorted keys):
  // down[0..3] (14 each), in_b, in_w, mid[0..1], out_b, out_w, up[0..3]
  const float* x     = F(0);
  const float* fenc  = F(1);
  const float* mask0 = F(2);
  Layer dn[4]  = { getL(3),  getL(17),  getL(31),  getL(45) };
  const float* in_b  = F(59);
  const float* in_w  = F(60);
  Layer md[2]  = { getL(61), getL(75) };
  const float* out_b = F(89);
  const float* out_w = F(90);
  Layer upl[4] = { getL(91), getL(105), getL(119), getL(133) };

  // workspace (~25.4 MB of floats)
  float* ws = (float*)d_ws;
  size_t off = 0;
  auto alloc = [&](size_t nf) { float* p = ws + off; off += nf; return p; };
  float* feat  = alloc((size_t)BB * CHN * TFULL);
  float* featB = alloc((size_t)BB * CHN * TFULL);
  float* ffb   = alloc((size_t)BB * CHN * TFULL);
  float* attm  = alloc((size_t)BB * CHN * TFULL);
  float* qbuf  = alloc((size_t)BB * EMB * TFULL);
  float* kbuf  = alloc((size_t)BB * EMB * TFULL);
  float* vbuf  = alloc((size_t)BB * EMB * TFULL);
  float* gbuf  = alloc((size_t)BB * EMB * TFULL);
  float* mAb   = alloc((size_t)BB * TFULL);
  float* mBb   = alloc((size_t)BB * TFULL);
  float* mu    = alloc(64);
  float* rs    = alloc(64);
  float* qwE   = alloc((size_t)EMB * CHN);
  float* qbE   = alloc(64);
  float* ffwT  = alloc((size_t)3 * CHN * CHN);

  auto pw = [&](int MT, const float* W, const float* bias, const float* X1, const float* X2,
                float a1, float a2, const float* res, const float* mr, float* Y,
                int Cin, int L) {
    dim3 g(L / 128, BB), blk(256);
    switch (MT) {
      case 2: pw_conv_wmma<2><<<g, blk, 0, stream>>>(W, bias, X1, X2, a1, a2, res, mr, Y, Cin, L); break;
      case 3: pw_conv_wmma<3><<<g, blk, 0, stream>>>(W, bias, X1, X2, a1, a2, res, mr, Y, Cin, L); break;
      default: pw_conv_wmma<4><<<g, blk, 0, stream>>>(W, bias, X1, X2, a1, a2, res, mr, Y, Cin, L); break;
    }
  };

  auto att_module = [&](int L, const Layer& P, const float* mr, int dil, int dr) {
    dim3 g(L / 128, BB), blk(256);
    // repack 3-tap weights, then ff = GELU(dilated conv3(feat))
    repack_w3_kernel<<<dim3(48), dim3(256), 0, stream>>>(P.ff_w, ffwT);
    dconv3_gelu_wmma<<<g, blk, 0, stream>>>(ffwT, P.ff_b, feat, ffb, L, dil);
    // BN stats + fold into q-conv weights
    bn_stats_kernel<<<dim3(CHN), dim3(256), 0, stream>>>(ffb, L, mu, rs);
    fold_bn_kernel<<<dim3(1), dim3(256), 0, stream>>>(P.q_w, P.q_b, P.bn_g, P.bn_b,
                                                      mu, rs, qwE, qbE);
    // q (B,32,L), k/v (B,32,T) from full-res encoder features
    pw(2, qwE, qbE, ffb, nullptr, 1.f, 0.f, nullptr, nullptr, qbuf, CHN, L);
    pw(2, P.k_w, P.k_b, fenc, nullptr, 1.f, 0.f, nullptr, nullptr, kbuf, CHN, TFULL);
    pw(2, P.v_w, P.v_b, fenc, nullptr, 1.f, 0.f, nullptr, nullptr, vbuf, CHN, TFULL);
    // sliding attention -> GELU output g
    dim3 ga(L / 256, BB);
    sliding_att_kernel<<<ga, dim3(256), 0, stream>>>(qbuf, kbuf, vbuf, mr, gbuf, L, dil, dil * dr);
    // ao conv (+bias) * mask
    pw(4, P.ao_w, P.ao_b, gbuf, nullptr, 1.f, 0.f, nullptr, mr, attm, EMB, L);
    // c1 conv of (ALPHA*att + ff), + residual feat, * mask  (in-place to feat)
    pw(4, P.c1_w, P.c1_b, attm, ffb, 1.0f, 1.0f, feat, mr, feat, CHN, L);
  };

  // input 1x1 conv
  pw(4, in_w, in_b, x, nullptr, 1.f, 0.f, nullptr, nullptr, feat, CHN, TFULL);

  const float* mr = mask0;
  float* mbuf[2] = { mAb, mBb };
  int mt = 0;
  int L = TFULL;

  // DOWN: att module, maxpool/2, mask[::2]
  for (int i = 0; i < 4; ++i) {
    att_module(L, dn[i], mr, 2, 1 << i);
    int Lh = L >> 1;
    int n1 = BB * CHN * Lh;
    maxpool2_kernel<<<dim3((n1 + 255) / 256), dim3(256), 0, stream>>>(feat, featB, n1, Lh);
    { float* t_ = feat; feat = featB; featB = t_; }
    int n2 = BB * Lh;
    rowsub2_kernel<<<dim3((n2 + 255) / 256), dim3(256), 0, stream>>>(mr, mbuf[mt], n2, Lh);
    mr = mbuf[mt]; mt ^= 1;
    L = Lh;
  }

  // MID
  for (int i = 0; i < 2; ++i) att_module(L, md[i], mr, 2, 16);

  // UP: repeat x2 (feat + mask), then att module
  const int updil[4] = { 8, 32, 128, 512 };
  const int updr[4]  = { 8, 4, 2, 1 };
  for (int i = 0; i < 4; ++i) {
    int L2 = L << 1;
    int n1 = BB * CHN * L2;
    rowup2_kernel<<<dim3((n1 + 255) / 256), dim3(256), 0, stream>>>(feat, featB, n1, L2);
    { float* t_ = feat; feat = featB; featB = t_; }
    int n2 = BB * L2;
    rowup2_kernel<<<dim3((n2 + 255) / 256), dim3(256), 0, stream>>>(mr, mbuf[mt], n2, L2);
    mr = mbuf[mt]; mt ^= 1;
    L = L2;
    att_module(L, upl[i], mr, updil[i], updr[i]);
  }

  // output conv (48 ch) * original mask
  pw(3, out_w, out_b, feat, nullptr, 1.f, 0.f, nullptr, mask0, (float*)d_out, CHN, TFULL);
}